// PretrainedGraphSAGEEncoder_37160057045125
// MI455X (gfx1250) — compile-verified
//
#include <hip/hip_runtime.h>
#include <hip/hip_bf16.h>

typedef __attribute__((ext_vector_type(2))) float v2f;
typedef __attribute__((ext_vector_type(8))) float v8f;

#define TILE_N 16
#define ROWS_PER_BLOCK 128   // 8 waves x 16 rows
#define MAX_K 256

// ---------------------------------------------------------------------------
// Degree: one thread per edge, atomic count into deg[dst].
// ---------------------------------------------------------------------------
__global__ __launch_bounds__(256) void deg_kernel(const long long* __restrict__ dstIdx,
                                                  float* __restrict__ deg, int E) {
    int e = blockIdx.x * blockDim.x + threadIdx.x;
    if (e < E) atomicAdd(&deg[(int)dstIdx[e]], 1.0f);
}

// deg[i] = 1 / max(deg[i], 1)  (in place -> becomes invdeg)
__global__ __launch_bounds__(256) void invdeg_kernel(float* __restrict__ deg, int Nn) {
    int i = blockIdx.x * blockDim.x + threadIdx.x;
    if (i < Nn) deg[i] = 1.0f / fmaxf(deg[i], 1.0f);
}

// ---------------------------------------------------------------------------
// Edge scatter: thread t handles (edge e, 4-float chunk c).
// msg[dst] += x[src]  (float atomics; msg + feature buffers are L2-resident)
// ---------------------------------------------------------------------------
__global__ __launch_bounds__(256) void scatter_kernel(const float* __restrict__ x,
                                                      const long long* __restrict__ srcIdx,
                                                      const long long* __restrict__ dstIdx,
                                                      float* __restrict__ msg,
                                                      int E, int d) {
    long long t = (long long)blockIdx.x * blockDim.x + threadIdx.x;
    int chunks = d >> 2;
    long long total = (long long)E * chunks;
    if (t >= total) return;
    int e = (int)(t / chunks);
    int c = (int)(t % chunks);
    int s  = (int)srcIdx[e];
    int dn = (int)dstIdx[e];
    const float4 v = *(const float4*)(x + (size_t)s * d + 4 * c);
    float* p = msg + (size_t)dn * d + 4 * c;
    atomicAdd(p + 0, v.x);
    atomicAdd(p + 1, v.y);
    atomicAdd(p + 2, v.z);
    atomicAdd(p + 3, v.w);
}

// ---------------------------------------------------------------------------
// Fused SAGE layer:  out = relu((A1 * invdeg[m]) @ B1 + A2 @ B2 + bias)
//   A1 = summed messages [M,K] (scaled to mean on the fly), B1 = W_l [K,N]
//   A2 = node features  [M,K],                              B2 = W_r [K,N]
// Block = 8 wave32, tile = 128 rows x 16 cols. W tiles staged in LDS
// (2 x K x 16 f32 <= 32 KB; stride-16 rows are LDS-bank-conflict free).
// V_WMMA_F32_16X16X4_F32 chained over both K streams into one f32 acc.
// Fragment layout (ISA 7.12.2, f32):
//   A 16x4 : lane = m (lane&15), lane-half h -> v2f {A[m][2h], A[m][2h+1]}
//   B 4x16 : lane = n (lane&15), lane-half h -> v2f {B[2h][n], B[2h+1][n]}
//   C/D    : VGPR r -> M = r + 8*h, N = lane&15
// M % 16 == 0 and per-wave row guard is uniform -> EXEC all-1s at every WMMA.
// ---------------------------------------------------------------------------
__global__ __launch_bounds__(256) void sage_gemm_relu(const float* __restrict__ A1,
                                                      const float* __restrict__ invdeg,
                                                      const float* __restrict__ B1,
                                                      const float* __restrict__ A2,
                                                      const float* __restrict__ B2,
                                                      const float* __restrict__ bias,
                                                      float* __restrict__ out,
                                                      int M, int K, int N) {
    __shared__ float Bs1[MAX_K * TILE_N];
    __shared__ float Bs2[MAX_K * TILE_N];

    const int m0 = blockIdx.x * ROWS_PER_BLOCK;
    const int n0 = blockIdx.y * TILE_N;

    // Cooperative stage of both weight column-tiles into LDS.
    for (int idx = threadIdx.x; idx < K * TILE_N; idx += 256) {
        int k = idx >> 4;
        int n = idx & 15;
        Bs1[idx] = B1[(size_t)k * N + n0 + n];
        Bs2[idx] = B2[(size_t)k * N + n0 + n];
    }
    __syncthreads();

    const int lane = threadIdx.x & 31;
    const int wave = threadIdx.x >> 5;
    const int mw = m0 + wave * 16;      // wave-uniform row base
    if (mw >= M) return;                 // whole wave exits together

    const int ml = lane & 15;
    const int kh = lane >> 4;
    const int m  = mw + ml;

    const float sc = invdeg[m];          // mean = sum * invdeg (folded in)
    const float bv = bias[n0 + ml];

    const float* a1 = A1 + (size_t)m * K + 2 * kh;
    const float* a2 = A2 + (size_t)m * K + 2 * kh;
    const float* bs1 = Bs1 + (size_t)(2 * kh) * TILE_N + ml;
    const float* bs2 = Bs2 + (size_t)(2 * kh) * TILE_N + ml;

    v8f acc = {};

    #pragma unroll 8
    for (int k = 0; k < K; k += 4) {
        v2f a = *(const v2f*)(a1 + k) * sc;
        v2f b;
        b.x = bs1[k * TILE_N];
        b.y = bs1[(k + 1) * TILE_N];
        acc = __builtin_amdgcn_wmma_f32_16x16x4_f32(false, a, false, b,
                                                    (short)0, acc, false, false);
    }

    #pragma unroll 8
    for (int k = 0; k < K; k += 4) {
        v2f a = *(const v2f*)(a2 + k);
        v2f b;
        b.x = bs2[k * TILE_N];
        b.y = bs2[(k + 1) * TILE_N];
        acc = __builtin_amdgcn_wmma_f32_16x16x4_f32(false, a, false, b,
                                                    (short)0, acc, false, false);
    }

    #pragma unroll
    for (int r = 0; r < 8; ++r) {
        int mr = mw + r + 8 * kh;
        float v = acc[r] + bv;
        out[(size_t)mr * N + n0 + ml] = v > 0.0f ? v : 0.0f;
    }
}

// ---------------------------------------------------------------------------
// Launch: 2-layer GraphSAGE (mean aggregation) forward.
// Inputs: 0:x[N,128] 1:edge_index[2,E](i64) 2:W1_l[128,256] 3:W1_r[128,256]
//         4:b1[256] 5:W2_l[256,256] 6:W2_r[256,256] 7:b2[256]
// ---------------------------------------------------------------------------
extern "C" void kernel_launch(void* const* d_in, const int* in_sizes, int n_in,
                              void* d_out, int out_size, void* d_ws, size_t ws_size,
                              hipStream_t stream) {
    const float*     x   = (const float*)d_in[0];
    const long long* ei  = (const long long*)d_in[1];
    const float*     W1l = (const float*)d_in[2];
    const float*     W1r = (const float*)d_in[3];
    const float*     b1  = (const float*)d_in[4];
    const float*     W2l = (const float*)d_in[5];
    const float*     W2r = (const float*)d_in[6];
    const float*     b2  = (const float*)d_in[7];

    const int Din = 128, H = 256;
    const int Nn = in_sizes[0] / Din;        // 50000
    const int E  = in_sizes[1] / 2;          // 800000
    const long long* srcIdx = ei;            // edge_index[0]
    const long long* dstIdx = ei + E;        // edge_index[1]

    // Workspace layout (floats): msg1[N,128] | msg2[N,256] | h1[N,256] | deg[N]
    float* ws   = (float*)d_ws;
    float* msg1 = ws;
    float* msg2 = msg1 + (size_t)Nn * Din;
    float* h1   = msg2 + (size_t)Nn * H;
    float* deg  = h1   + (size_t)Nn * H;

    hipMemsetAsync(msg1, 0, (size_t)Nn * Din * sizeof(float), stream);
    hipMemsetAsync(msg2, 0, (size_t)Nn * H   * sizeof(float), stream);
    hipMemsetAsync(deg,  0, (size_t)Nn * sizeof(float), stream);

    // Degrees (shared by both layers), then invert in place.
    deg_kernel<<<(E + 255) / 256, 256, 0, stream>>>(dstIdx, deg, E);
    invdeg_kernel<<<(Nn + 255) / 256, 256, 0, stream>>>(deg, Nn);

    // Layer 1 aggregation: msg1 += x[src] over edges.
    {
        long long tot = (long long)E * (Din / 4);
        scatter_kernel<<<(unsigned)((tot + 255) / 256), 256, 0, stream>>>(
            x, srcIdx, dstIdx, msg1, E, Din);
    }

    // Layer 1: h1 = relu((msg1*invdeg) @ W1_l + x @ W1_r + b1)
    {
        dim3 grid((Nn + ROWS_PER_BLOCK - 1) / ROWS_PER_BLOCK, H / TILE_N);
        sage_gemm_relu<<<grid, 256, 0, stream>>>(
            msg1, deg, W1l, x, W1r, b1, h1, Nn, Din, H);
    }

    // Layer 2 aggregation: msg2 += h1[src] over edges.
    {
        long long tot = (long long)E * (H / 4);
        scatter_kernel<<<(unsigned)((tot + 255) / 256), 256, 0, stream>>>(
            h1, srcIdx, dstIdx, msg2, E, H);
    }

    // Layer 2: out = relu((msg2*invdeg) @ W2_l + h1 @ W2_r + b2)
    {
        dim3 grid((Nn + ROWS_PER_BLOCK - 1) / ROWS_PER_BLOCK, H / TILE_N);
        sage_gemm_relu<<<grid, 256, 0, stream>>>(
            msg2, deg, W2l, h1, W2r, b2, (float*)d_out, Nn, H, H);
    }
}